// ControlValLoss_21303037788495
// MI455X (gfx1250) — compile-verified
//
#include <hip/hip_runtime.h>
#include <stdint.h>

// Problem constants (from reference)
#define TOKENS        16320   // B * (S-2) = 32 * 510
#define VOCAB         2004
#define PAD_TOK       2003
#define S_FULL        512
#define S_USE         510
#define VEC4_PER_ROW  501     // 2004 / 4
#define CHUNKS        16      // ceil(501 / 32)
#define WAVES_PER_BLK 8

// ---------------- CDNA5 async global->LDS helpers (gfx1250) ----------------
// Per-lane async copy of 16 bytes from global memory to LDS.
// VDST (first operand) = 32-bit LDS byte address, VADDR = 64-bit global addr.
__device__ __forceinline__ void async_ld16(uint32_t lds_addr, uint64_t gaddr) {
    asm volatile("global_load_async_to_lds_b128 %0, %1, off"
                 :: "v"(lds_addr), "v"(gaddr)
                 : "memory");
}
__device__ __forceinline__ void wait_async_le1() {
    asm volatile("s_wait_asynccnt 0x1" ::: "memory");
}
__device__ __forceinline__ void wait_async_le0() {
    asm volatile("s_wait_asynccnt 0x0" ::: "memory");
}

// ---------------- Kernel 1: per-token argmax over 2004 logits ----------------
// One wave32 per token row; double-buffered async-to-LDS streaming.
__global__ __launch_bounds__(256)
void tok_argmax_kernel(const float* __restrict__ pred, int* __restrict__ tok_out) {
    __shared__ float4 buf[WAVES_PER_BLK][2][32];   // 8 KB total

    const int wave  = threadIdx.x >> 5;
    const int lane  = threadIdx.x & 31;
    const int token = blockIdx.x * WAVES_PER_BLK + wave;
    if (token >= TOKENS) return;

    const int b = token / S_USE;
    const int s = token - b * S_USE;
    // pred is (B, S_FULL, VOCAB); we read row (b, s) with s < 510
    const float4* row =
        (const float4*)(pred + (size_t)(b * S_FULL + s) * VOCAB);

    // LDS byte addresses for this lane's two buffer slots (low 32 bits of the
    // generic pointer are the LDS offset on gfx1250).
    const uint32_t lds0 = (uint32_t)(uintptr_t)&buf[wave][0][lane];
    const uint32_t lds1 = (uint32_t)(uintptr_t)&buf[wave][1][lane];

    // Prologue: chunk 0 (indices lane < 501 always valid)
    async_ld16(lds0, (uint64_t)(uintptr_t)(row + lane));

    float bestV = -3.0e38f;
    int   bestI = 0;

    for (int c = 0; c < CHUNKS; ++c) {
        const int nextIdx = (c + 1) * 32 + lane;
        if (c + 1 < CHUNKS) {
            if (nextIdx < VEC4_PER_ROW) {
                async_ld16(((c + 1) & 1) ? lds1 : lds0,
                           (uint64_t)(uintptr_t)(row + nextIdx));
            }
            wait_async_le1();   // chunk c has landed in LDS
        } else {
            wait_async_le0();   // drain
        }
        const int idx = c * 32 + lane;
        if (idx < VEC4_PER_ROW) {
            const float4 v = buf[wave][c & 1][lane];
            const int base = idx * 4;
            if (v.x > bestV) { bestV = v.x; bestI = base;     }
            if (v.y > bestV) { bestV = v.y; bestI = base + 1; }
            if (v.z > bestV) { bestV = v.z; bestI = base + 2; }
            if (v.w > bestV) { bestV = v.w; bestI = base + 3; }
        }
    }

    // Wave32 argmax reduction, first-occurrence tie-break (matches jnp.argmax)
    for (int off = 16; off > 0; off >>= 1) {
        const float ov = __shfl_xor(bestV, off, 32);
        const int   oi = __shfl_xor(bestI, off, 32);
        if (ov > bestV || (ov == bestV && oi < bestI)) { bestV = ov; bestI = oi; }
    }
    if (lane == 0) tok_out[token] = bestI;
}

// ---------------- Kernel 2: mask scan + compacted gather + loss ----------------
__global__ __launch_bounds__(256)
void loss_finalize_kernel(const int* __restrict__ tok,
                          const float* __restrict__ gt_acc,
                          const float* __restrict__ gt_steer,
                          const int* __restrict__ gt_rev,
                          float* __restrict__ out) {
    __shared__ int   scan[256];
    __shared__ float r_acc, r_steer, r_rev, r_cnt, r_ce;

    const int t     = threadIdx.x;
    const int start = t * 64;
    const int end   = min(start + 64, TOKENS);

    // Pass 1: local mask count
    int cnt = 0;
    for (int i = start; i < end; ++i) cnt += (tok[i] != PAD_TOK) ? 1 : 0;
    scan[t] = cnt;
    __syncthreads();

    // Blockwide inclusive scan (Hillis-Steele)
    for (int off = 1; off < 256; off <<= 1) {
        const int v = (t >= off) ? scan[t - off] : 0;
        __syncthreads();
        scan[t] += v;
        __syncthreads();
    }
    int run = scan[t] - cnt;   // exclusive prefix = pos for first masked token

    if (t == 0) { r_acc = 0.f; r_steer = 0.f; r_rev = 0.f; r_cnt = 0.f; r_ce = 0.f; }
    __syncthreads();

    const float scale = 1.0f / 9.0f;                 // 1/(BASE-1)
    const float lse   = 1.31326168751822286f;        // log1p(exp(1))

    float accS = 0.f, steerS = 0.f, revS = 0.f, myCnt = 0.f, myCe = 0.f;

    // Pass 2: decode + compacted gather + loss terms (unmasked tokens
    // contribute exactly 0 in the reference, so we skip them).
    for (int i = start; i < end; ++i) {
        const int tk = tok[i];
        if (tk == PAD_TOK) continue;
        const int p = run++;

        const int rev          = tk & 1;
        const int steer_tok    = (tk >> 1) % 10;
        const int brake_tok    = (tk / 20) % 10;
        const int throttle_tok = (tk / 200) % 10;
        const float acc_p   = (float)(throttle_tok - brake_tok) * scale;
        const float steer_p = (float)steer_tok * scale * 2.0f - 1.0f;

        const float d1 = acc_p - gt_acc[p];
        const float a1 = fabsf(d1);
        accS += (a1 < 1.0f) ? 0.5f * d1 * d1 : a1 - 0.5f;

        const float d2 = steer_p - gt_steer[p];
        const float a2 = fabsf(d2);
        steerS += (a2 < 1.0f) ? 0.5f * d2 * d2 : a2 - 0.5f;

        myCnt += 1.0f;

        const int gr  = gt_rev[p];
        const float ce = (gr != PAD_TOK) ? 1.0f : 0.0f;
        myCe += ce;
        const float logit_at_gt = (rev == gr) ? 1.0f : 0.0f;
        revS += (lse - logit_at_gt) * ce;
    }

    atomicAdd(&r_acc,   accS);
    atomicAdd(&r_steer, steerS);
    atomicAdd(&r_rev,   revS);
    atomicAdd(&r_cnt,   myCnt);
    atomicAdd(&r_ce,    myCe);
    __syncthreads();

    if (t == 0) {
        const float c  = fmaxf(r_cnt, 1.0f);
        const float cc = fmaxf(r_ce, 1.0f);
        out[0] = r_acc / c + r_steer / c;   // acc_steer_val_loss
        out[1] = r_rev / cc;                // reverse_val_loss
    }
}

// ---------------- Launch ----------------
extern "C" void kernel_launch(void* const* d_in, const int* in_sizes, int n_in,
                              void* d_out, int out_size, void* d_ws, size_t ws_size,
                              hipStream_t stream) {
    const float* pred     = (const float*)d_in[0];
    const float* gt_acc   = (const float*)d_in[1];
    const float* gt_steer = (const float*)d_in[2];
    const int*   gt_rev   = (const int*)d_in[3];
    float* out = (float*)d_out;
    int* tok_ws = (int*)d_ws;   // 16320 ints of scratch

    tok_argmax_kernel<<<TOKENS / WAVES_PER_BLK, 256, 0, stream>>>(pred, tok_ws);
    loss_finalize_kernel<<<1, 256, 0, stream>>>(tok_ws, gt_acc, gt_steer, gt_rev, out);
}